// lowfid_17446157156789
// MI455X (gfx1250) — compile-verified
//
#include <hip/hip_runtime.h>

#define Bn 8
#define Hn 128
#define Wn 128
#define NTHREADS 1024

typedef unsigned int u32x4 __attribute__((ext_vector_type(4)));
typedef int          i32x4 __attribute__((ext_vector_type(4)));
typedef int          i32x8 __attribute__((ext_vector_type(8)));

#ifndef __has_builtin
#define __has_builtin(x) 0
#endif

#if defined(__AMDGCN__) && __has_builtin(__builtin_amdgcn_tensor_load_to_lds) && __has_builtin(__builtin_amdgcn_s_wait_tensorcnt)
#define USE_TDM 1
#else
#define USE_TDM 0
#endif

// One workgroup (32 wave32s) per batch image; solve is fully LDS-resident.
//   bufA [0      .. 65536)  : staged K (TDM), then ping buffer
//   bufB [65536  .. 131072) : T0, then pong buffer
//   kRow [131072 .. 131584) : saved K row 64 for the flux
//   red  [131584 .. 132096) : reduction scratch
// Thread tile: 8 rows x 2 adjacent columns -> b64 LDS traffic, half the DS ops.
extern "C" __global__ __launch_bounds__(NTHREADS)
void jacobi_fused(const float* __restrict__ cond,
                  const int*   __restrict__ itersPtr,
                  float*       __restrict__ out)
{
    extern __shared__ float smem[];
    float* bufA = smem;
    float* bufB = smem + Hn * Wn;
    float* kRow = smem + 2 * Hn * Wn;
    float* red  = smem + 2 * Hn * Wn + Wn;

    const int tid = threadIdx.x;
    const int b   = blockIdx.x;
    const int xi  = tid & 63;          // column-pair index, lane-consecutive
    const int x2  = xi << 1;           // even column of the pair
    const int ty  = tid >> 6;          // 0..15, 8 rows per thread
    const int y0  = ty << 3;

    // clamped halo columns (edge replication falls out of address clamping)
    const int xm = (x2 == 0) ? 0 : x2 - 1;
    const int xp = (x2 + 2 > Wn - 1) ? (Wn - 1) : x2 + 2;

    const float* gk = cond + (size_t)b * Hn * Wn;

    // ---- Stage conductivity tile Global -> LDS via Tensor Data Mover ----
#if USE_TDM
    if (tid < 32) {  // wave 0 issues the DMA
        unsigned long long ga = (unsigned long long)(void*)gk;
        unsigned ldsOff = (unsigned)(unsigned long long)(void*)bufA;
        u32x4 g0;
        g0.x = 1u;                                             // count=1 (valid)
        g0.y = ldsOff;                                         // lds_addr [63:32]
        g0.z = (unsigned)(ga & 0xFFFFFFFFull);                 // global_addr[31:0]
        g0.w = (unsigned)((ga >> 32) & 0x1FFFFFFull)           // global_addr[56:32]
               | (2u << 30);                                   // type=2 ("image")
        i32x8 g1;
        g1[0] = 0x00020000;        // data_size=2 (4 bytes)
        g1[1] = Wn << 16;          // tensor_dim0 = 128
        g1[2] = Hn << 16;          // tensor_dim1 = 128
        g1[3] = Wn << 16;          // tile_dim0   = 128
        g1[4] = Hn;                // tile_dim1   = 128, tile_dim2 = 0
        g1[5] = Wn;                // tensor_dim0_stride = 128
        g1[6] = (int)0x40000000;   // tensor_dim1_stride = 16384 (low 16 bits)
        g1[7] = 0;
        i32x4 z4 = {0, 0, 0, 0};
#if __clang_major__ >= 23
        i32x8 z8 = {0, 0, 0, 0, 0, 0, 0, 0};
        __builtin_amdgcn_tensor_load_to_lds(g0, g1, z4, z4, z8, 0);
#else
        __builtin_amdgcn_tensor_load_to_lds(g0, g1, z4, z4, 0);
#endif
        __builtin_amdgcn_s_wait_tensorcnt(0);
    }
#else
    for (int i = tid; i < Hn * Wn; i += NTHREADS) bufA[i] = gk[i];
#endif
    __syncthreads();

    // ---- Precompute per-cell coefficients cX = kX/denom into VGPRs ----
    float cNx[8], cNy[8], cSx[8], cSy[8], cWx[8], cWy[8], cEx[8], cEy[8];
#pragma unroll
    for (int i = 0; i < 8; ++i) {
        const int y  = y0 + i;
        const int yn = (y == 0)      ? 0      : y - 1;
        const int ys = (y == Hn - 1) ? Hn - 1 : y + 1;
        const float2 kc = *(const float2*)&bufA[y  * Wn + x2];
        const float2 kn = *(const float2*)&bufA[yn * Wn + x2];
        const float2 ks = *(const float2*)&bufA[ys * Wn + x2];
        const float  kw = bufA[y * Wn + xm];   // clamped -> edge replication
        const float  ke = bufA[y * Wn + xp];
        const float kN0 = 0.5f * (kc.x + kn.x), kN1 = 0.5f * (kc.y + kn.y);
        const float kS0 = 0.5f * (kc.x + ks.x), kS1 = 0.5f * (kc.y + ks.y);
        const float kW0 = 0.5f * (kc.x + kw),   kW1 = 0.5f * (kc.y + kc.x);
        const float kE0 = 0.5f * (kc.x + kc.y), kE1 = 0.5f * (kc.y + ke);
        const float i0 = 1.0f / (kN0 + kS0 + kW0 + kE0);
        const float i1 = 1.0f / (kN1 + kS1 + kW1 + kE1);
        cNx[i] = kN0 * i0; cNy[i] = kN1 * i1;
        cSx[i] = kS0 * i0; cSy[i] = kS1 * i1;
        cWx[i] = kW0 * i0; cWy[i] = kW1 * i1;
        cEx[i] = kE0 * i0; cEy[i] = kE1 * i1;
    }
    if (ty == 8) {                       // rows 64..71 group: save K row 64
        kRow[x2]     = bufA[64 * Wn + x2];
        kRow[x2 + 1] = bufA[64 * Wn + x2 + 1];
    }

    // ---- T0: zeros, top row = 1 ----
#pragma unroll
    for (int i = 0; i < 8; ++i) {
        const int y = y0 + i;
        float2 t0; t0.x = (y == 0) ? 1.0f : 0.0f; t0.y = t0.x;
        *(float2*)&bufB[y * Wn + x2] = t0;
    }
    __syncthreads();   // coeff reads of bufA done; T0 visible

    // ---- Jacobi sweeps: LDS ping-pong, one barrier per sweep ----
    const int iters = itersPtr[0];
    float* src = bufB;
    float* dst = bufA;
    for (int it = 0; it < iters; ++it) {
        // rolling column pairs: tN2 = T[y-1][x2..x2+1], tC2 = T[y][x2..x2+1]
        float2 tN2 = *(const float2*)&src[((y0 == 0) ? 0 : (y0 - 1)) * Wn + x2];
        float2 tC2 = *(const float2*)&src[y0 * Wn + x2];
#pragma unroll
        for (int i = 0; i < 8; ++i) {
            const int y   = y0 + i;
            const int ysr = (y == Hn - 1) ? (Hn - 1) : (y + 1);
            const float2 tS2 = *(const float2*)&src[ysr * Wn + x2]; // b64
            const float  wL  = src[y * Wn + xm];   // halo pair: mergeable into
            const float  eR  = src[y * Wn + xp];   //   one ds_load_2addr_b32
            // interior W/E neighbors come from the rolling center pair
            float v0 = __builtin_fmaf(cNx[i], tN2.x,
                       __builtin_fmaf(cSx[i], tS2.x,
                       __builtin_fmaf(cWx[i], wL,    cEx[i] * tC2.y)));
            float v1 = __builtin_fmaf(cNy[i], tN2.y,
                       __builtin_fmaf(cSy[i], tS2.y,
                       __builtin_fmaf(cWy[i], tC2.x, cEy[i] * eR)));
            if (y == 0)      { v0 = 1.0f; v1 = 1.0f; }  // Dirichlet top
            if (y == Hn - 1) { v0 = 0.0f; v1 = 0.0f; }  // Dirichlet bottom
            float2 v2; v2.x = v0; v2.y = v1;
            *(float2*)&dst[y * Wn + x2] = v2;            // b64 store
            tN2 = tC2;
            tC2 = tS2;
        }
        __syncthreads();
        float* t = src; src = dst; dst = t;
    }
    // final T is in `src`

    // ---- Flux across mid-row: kappa = sum_x -k[64][x]*(T[65][x]-T[64][x]) ----
    if (tid < Wn) {
        const float t64 = src[64 * Wn + tid];
        const float t65 = src[65 * Wn + tid];
        red[tid] = -kRow[tid] * (t65 - t64);
    }
    __syncthreads();
#pragma unroll
    for (int s = Wn / 2; s > 0; s >>= 1) {
        if (tid < s) red[tid] += red[tid + s];
        __syncthreads();
    }
    if (tid == 0) out[b] = red[0];
}

extern "C" void kernel_launch(void* const* d_in, const int* in_sizes, int n_in,
                              void* d_out, int out_size, void* d_ws, size_t ws_size,
                              hipStream_t stream) {
    const float* cond  = (const float*)d_in[0];
    const int*   iters = (const int*)d_in[1];
    float*       out   = (float*)d_out;
    (void)in_sizes; (void)n_in; (void)out_size; (void)d_ws; (void)ws_size;

    const size_t shmem = (size_t)(2 * Hn * Wn + 2 * Wn) * sizeof(float); // 132096 B < 320 KB/WGP
    (void)hipFuncSetAttribute((const void*)jacobi_fused,
                              hipFuncAttributeMaxDynamicSharedMemorySize,
                              (int)shmem);
    jacobi_fused<<<dim3(Bn), dim3(NTHREADS), shmem, stream>>>(cond, iters, out);
}